// GCN_59657095741934
// MI455X (gfx1250) — compile-verified
//
#include <hip/hip_runtime.h>
#include <hip/hip_bf16.h>

typedef __attribute__((ext_vector_type(2))) float v2f;
typedef __attribute__((ext_vector_type(8))) float v8f;

// ---------------------------------------------------------------------------
// Zero-fill kernel (grid-stride). d_ws is poisoned 0xAA, so pooled buffers
// must be zeroed before the atomic-max scatter each layer.
// ---------------------------------------------------------------------------
__global__ void sage_zero_kernel(float* __restrict__ p, long long n) {
    long long i = (long long)blockIdx.x * blockDim.x + threadIdx.x;
    long long stride = (long long)gridDim.x * blockDim.x;
    for (; i < n; i += stride) p[i] = 0.0f;
}

// ---------------------------------------------------------------------------
// XP = relu(X @ W + b)   X:[nrows x 64], W:[64 x 64] row-major, XP:[nrows x 64]
// One wave computes a 16-row x 64-col tile via V_WMMA_F32_16X16X4_F32.
// W is staged in LDS as row-pairs so each B fragment is one ds_load_b64:
//   sW[p*64 + c] = { W[2p][c], W[2p+1][c] },  lane pair index = k0/2 + lhalf.
// ---------------------------------------------------------------------------
__global__ void sage_pool_gemm_kernel(const float* __restrict__ X,
                                      const float* __restrict__ W,
                                      const float* __restrict__ b,
                                      float* __restrict__ XP, int nrows) {
    __shared__ float2 sW[32 * 64];                      // 16 KB

    // Stage W -> LDS, packing consecutive K rows into float2 pairs.
    for (int i = threadIdx.x; i < 32 * 64; i += blockDim.x) {
        const int p = i >> 6, c = i & 63;
        float2 t;
        t.x = W[(2 * p) * 64 + c];
        t.y = W[(2 * p + 1) * 64 + c];
        sW[i] = t;
    }
    __syncthreads();

    const int wave = threadIdx.x >> 5;                  // 4 waves / block
    const int lane = threadIdx.x & 31;
    const int rowbase = (blockIdx.x * (blockDim.x >> 5) + wave) * 16;
    const int lhalf = lane >> 4;                        // 0: lanes 0-15, 1: lanes 16-31
    const int l16 = lane & 15;
    const int arow = min(rowbase + l16, nrows - 1);     // clamped load row (EXEC stays full)
    const float* __restrict__ xr = X + (long long)arow * 64;

    v8f c[4] = {};
    for (int k0 = 0; k0 < 64; k0 += 4) {
        const int ka = k0 + 2 * lhalf;                  // A: lanes16-31 hold K+2..K+3
        const int kp = (k0 >> 1) + lhalf;               // LDS pair row
        v2f a;
        a.x = xr[ka];
        a.y = xr[ka + 1];
#pragma unroll
        for (int t = 0; t < 4; ++t) {
            const float2 w = sW[kp * 64 + t * 16 + l16];
            v2f bb; bb.x = w.x; bb.y = w.y;
            c[t] = __builtin_amdgcn_wmma_f32_16x16x4_f32(
                false, a, false, bb, (short)0, c[t], false, false);
        }
    }

    // Epilogue: wave-uniform fast path for full tiles; bias hoisted per tile.
    if (rowbase + 16 <= nrows) {
#pragma unroll
        for (int t = 0; t < 4; ++t) {
            const int col = t * 16 + l16;
            const float bv = b[col];
            float* __restrict__ yp = XP + (long long)(rowbase + 8 * lhalf) * 64 + col;
#pragma unroll
            for (int j = 0; j < 8; ++j)
                yp[j * 64] = fmaxf(c[t][j] + bv, 0.0f);
        }
    } else {
#pragma unroll
        for (int t = 0; t < 4; ++t) {
            const int col = t * 16 + l16;
            const float bv = b[col];
#pragma unroll
            for (int j = 0; j < 8; ++j) {
                const int row = rowbase + j + 8 * lhalf;
                if (row < nrows)
                    XP[(long long)row * 64 + col] = fmaxf(c[t][j] + bv, 0.0f);
            }
        }
    }
}

// ---------------------------------------------------------------------------
// Segment-max scatter: pooled[dst] = max(pooled[dst], XP[src]) elementwise.
// XP >= 0 (post-ReLU) so IEEE bit patterns are monotonic under u32 compare ->
// GLOBAL_ATOMIC_MAX_U32 implements float max exactly; pooled pre-zeroed also
// reproduces the reference's empty-segment -> 0 behavior.
// 16 lanes per edge, 4 floats per lane (one 16B load, four u32 atomics).
// ---------------------------------------------------------------------------
__global__ void sage_edge_max_kernel(const int* __restrict__ src,
                                     const int* __restrict__ dst,
                                     const float* __restrict__ XP,
                                     unsigned int* __restrict__ pooled,
                                     int nedges) {
    const long long t = (long long)blockIdx.x * blockDim.x + threadIdx.x;
    const long long e = t >> 4;
    const int sub = (int)(t & 15);
    if (e >= nedges) return;
    const int s = src[e];
    const int d = dst[e];
    const uint4 v = *(const uint4*)(XP + (long long)s * 64 + sub * 4);
    unsigned int* pp = pooled + (long long)d * 64 + sub * 4;
    atomicMax(pp + 0, v.x);
    atomicMax(pp + 1, v.y);
    atomicMax(pp + 2, v.z);
    atomicMax(pp + 3, v.w);
}

// ---------------------------------------------------------------------------
// Y = act(X @ Wself + P @ Wneigh + bias)
// NTILES = NOUT/16 (4 for width-64 hidden, 1 for width-16 logits).
// LEAKY: apply leaky_relu(0.01) (layer-1 output); else identity (final logits).
// Both GEMMs accumulate into the same WMMA accumulators; both weight matrices
// staged in LDS as row-pairs (one ds_load_b64 per B fragment).
// ---------------------------------------------------------------------------
template <int NTILES, bool LEAKY>
__global__ void sage_out_gemm_kernel(const float* __restrict__ X,
                                     const float* __restrict__ Wself,
                                     const float* __restrict__ P,
                                     const float* __restrict__ Wneigh,
                                     const float* __restrict__ bias,
                                     float* __restrict__ Y, int nrows) {
    const int NOUT = NTILES * 16;
    __shared__ float2 sWs[32 * NTILES * 16];
    __shared__ float2 sWn[32 * NTILES * 16];

    for (int i = threadIdx.x; i < 32 * NOUT; i += blockDim.x) {
        const int p = i / NOUT, c = i % NOUT;
        float2 t0, t1;
        t0.x = Wself[(2 * p) * NOUT + c];
        t0.y = Wself[(2 * p + 1) * NOUT + c];
        t1.x = Wneigh[(2 * p) * NOUT + c];
        t1.y = Wneigh[(2 * p + 1) * NOUT + c];
        sWs[i] = t0;
        sWn[i] = t1;
    }
    __syncthreads();

    const int wave = threadIdx.x >> 5;
    const int lane = threadIdx.x & 31;
    const int rowbase = (blockIdx.x * (blockDim.x >> 5) + wave) * 16;
    const int lhalf = lane >> 4;
    const int l16 = lane & 15;
    const int arow = min(rowbase + l16, nrows - 1);
    const float* __restrict__ xr = X + (long long)arow * 64;
    const float* __restrict__ pr = P + (long long)arow * 64;

    v8f c[NTILES] = {};
    for (int k0 = 0; k0 < 64; k0 += 4) {
        const int ka = k0 + 2 * lhalf;
        const int kp = (k0 >> 1) + lhalf;
        v2f a1, a2;
        a1.x = xr[ka];  a1.y = xr[ka + 1];
        a2.x = pr[ka];  a2.y = pr[ka + 1];
#pragma unroll
        for (int t = 0; t < NTILES; ++t) {
            const float2 ws = sWs[kp * NOUT + t * 16 + l16];
            const float2 wn = sWn[kp * NOUT + t * 16 + l16];
            v2f b1; b1.x = ws.x; b1.y = ws.y;
            v2f b2; b2.x = wn.x; b2.y = wn.y;
            c[t] = __builtin_amdgcn_wmma_f32_16x16x4_f32(
                false, a1, false, b1, (short)0, c[t], false, false);
            c[t] = __builtin_amdgcn_wmma_f32_16x16x4_f32(
                false, a2, false, b2, (short)0, c[t], false, false);
        }
    }

    if (rowbase + 16 <= nrows) {
#pragma unroll
        for (int t = 0; t < NTILES; ++t) {
            const int col = t * 16 + l16;
            const float bv = bias[col];
            float* __restrict__ yp = Y + (long long)(rowbase + 8 * lhalf) * NOUT + col;
#pragma unroll
            for (int j = 0; j < 8; ++j) {
                float v = c[t][j] + bv;
                if (LEAKY) v = (v >= 0.0f) ? v : 0.01f * v;
                yp[j * NOUT] = v;
            }
        }
    } else {
#pragma unroll
        for (int t = 0; t < NTILES; ++t) {
            const int col = t * 16 + l16;
            const float bv = bias[col];
#pragma unroll
            for (int j = 0; j < 8; ++j) {
                const int row = rowbase + j + 8 * lhalf;
                if (row < nrows) {
                    float v = c[t][j] + bv;
                    if (LEAKY) v = (v >= 0.0f) ? v : 0.01f * v;
                    Y[(long long)row * NOUT + col] = v;
                }
            }
        }
    }
}

extern "C" void kernel_launch(void* const* d_in, const int* in_sizes, int n_in,
                              void* d_out, int out_size, void* d_ws, size_t ws_size,
                              hipStream_t stream) {
    // setup_inputs() order:
    const float* in_feat  = (const float*)d_in[0];
    const int*   src      = (const int*)d_in[1];
    const int*   dst      = (const int*)d_in[2];
    const float* W_pool1  = (const float*)d_in[3];
    const float* b_pool1  = (const float*)d_in[4];
    const float* W_self1  = (const float*)d_in[5];
    const float* W_neigh1 = (const float*)d_in[6];
    const float* bias1    = (const float*)d_in[7];
    const float* W_pool2  = (const float*)d_in[8];
    const float* b_pool2  = (const float*)d_in[9];
    const float* W_self2  = (const float*)d_in[10];
    const float* W_neigh2 = (const float*)d_in[11];
    const float* bias2    = (const float*)d_in[12];

    const int N = in_sizes[0] / 64;   // 100000
    const int E = in_sizes[1];        // 1600000
    const long long NF = (long long)N * 64;

    // Workspace: XP (reused for both layers), P (pooled, reused), H (hidden).
    float* XP = (float*)d_ws;
    float* P  = XP + NF;
    float* H  = P + NF;

    const int gemm_threads = 128;                       // 4 waves -> 64 rows / block
    const int gemm_blocks  = (N + 63) / 64;
    int zero_blocks = (int)((NF + 255) / 256);
    if (zero_blocks > 4096) zero_blocks = 4096;
    const long long edge_threads_total = (long long)E * 16;
    const int edge_blocks = (int)((edge_threads_total + 255) / 256);

    // ---- Layer 1 ----
    sage_zero_kernel<<<zero_blocks, 256, 0, stream>>>(P, NF);
    sage_pool_gemm_kernel<<<gemm_blocks, gemm_threads, 0, stream>>>(
        in_feat, W_pool1, b_pool1, XP, N);
    sage_edge_max_kernel<<<edge_blocks, 256, 0, stream>>>(
        src, dst, XP, (unsigned int*)P, E);
    sage_out_gemm_kernel<4, true><<<gemm_blocks, gemm_threads, 0, stream>>>(
        in_feat, W_self1, P, W_neigh1, bias1, H, N);

    // ---- Layer 2 ----
    sage_pool_gemm_kernel<<<gemm_blocks, gemm_threads, 0, stream>>>(
        H, W_pool2, b_pool2, XP, N);
    sage_zero_kernel<<<zero_blocks, 256, 0, stream>>>(P, NF);
    sage_edge_max_kernel<<<edge_blocks, 256, 0, stream>>>(
        src, dst, XP, (unsigned int*)P, E);
    sage_out_gemm_kernel<1, false><<<gemm_blocks, gemm_threads, 0, stream>>>(
        H, W_self2, P, W_neigh2, bias2, (float*)d_out, N);
}